// RecommendationModel_32040456028763
// MI455X (gfx1250) — compile-verified
//
#include <hip/hip_runtime.h>
#include <cmath>
#include <cstdint>

#define HID 128

typedef __attribute__((ext_vector_type(2))) float v2f;
typedef __attribute__((ext_vector_type(8))) float v8f;

// ---------------------------------------------------------------- utilities
__global__ __launch_bounds__(256) void fill_f32(float* p, float v, int n) {
    int i = blockIdx.x * 256 + threadIdx.x;
    if (i < n) p[i] = v;
}

__device__ inline void atomicMaxFloat(float* addr, float val) {
    // monotonic toward max; valid with -inf init, mixed sign writers
    if (val >= 0.0f) atomicMax((int*)addr, __float_as_int(val));
    else             atomicMin((unsigned int*)addr, __float_as_uint(val));
}

// ---------------------------------------------------------------- batchnorm
__global__ __launch_bounds__(256) void bn_stats(const float* __restrict__ x, int n,
                                                float* __restrict__ sums) {
    __shared__ float s[256], q[256];
    int tid = threadIdx.x;
    int c = tid & 127, half = tid >> 7;
    int r0 = blockIdx.x * 64;
    int rend = min(r0 + 64, n);
    float ls = 0.f, lq = 0.f;
    for (int r = r0 + half; r < rend; r += 2) {
        float v = x[(size_t)r * HID + c];
        ls += v; lq += v * v;
    }
    s[tid] = ls; q[tid] = lq;
    __syncthreads();
    if (tid < 128) {
        atomicAdd(&sums[c],        s[tid] + s[tid + 128]);
        atomicAdd(&sums[128 + c],  q[tid] + q[tid + 128]);
    }
}

__global__ void bn_coef(const float* __restrict__ sums, int n,
                        const float* __restrict__ g, const float* __restrict__ b,
                        float* __restrict__ coef) {
    int c = threadIdx.x;  // 128
    float inv = 1.0f / (float)n;
    float mu = sums[c] * inv;
    float var = sums[128 + c] * inv - mu * mu;          // biased var (reference)
    float A = rsqrtf(var + 1e-5f) * g[c];
    coef[c] = A;
    coef[128 + c] = b[c] - mu * A;
}

__global__ __launch_bounds__(256) void bn_apply(const float* __restrict__ x,
                                                const float* __restrict__ coef,
                                                float* __restrict__ y, int n) {
    size_t i = (size_t)blockIdx.x * 256 + threadIdx.x;
    if (i >= (size_t)n * HID) return;
    int c = (int)(i & 127);
    y[i] = x[i] * coef[c] + coef[128 + c];
}

// ------------------------------------------------- fold rel matrix into W
// Weff[m][j] = sum_d W[h*64+d][j] * rel[h][d][e],  m = h*64+e
// beff[m]    = sum_d b[h*64+d]    * rel[h][d][e]
__global__ __launch_bounds__(128) void fold_wmat(const float* __restrict__ W,
                                                 const float* __restrict__ b,
                                                 const float* __restrict__ rel,
                                                 float* __restrict__ Weff,
                                                 float* __restrict__ beff) {
    int m = blockIdx.x;     // 0..127
    int j = threadIdx.x;    // 0..127
    int h = m >> 6, e = m & 63;
    const float* rh = rel + h * 4096 + e;   // rel[h][d][e] = rh[d*64]
    float acc = 0.f;
    for (int d = 0; d < 64; ++d)
        acc += W[(size_t)(h * 64 + d) * HID + j] * rh[d * 64];
    Weff[(size_t)m * HID + j] = acc;
    if (j == 0) {
        float bb = 0.f;
        for (int d = 0; d < 64; ++d) bb += b[h * 64 + d] * rh[d * 64];
        beff[m] = bb;
    }
}

// ------------------------------------------------------------- WMMA GEMM
// Y[N x 128] = X[N x 128] @ W[128 x 128]^T + bias   (fp32 V_WMMA_F32_16X16X4_F32)
// 256 threads = 8 waves; wave w owns rows [w*16, w*16+16), all 128 output cols
// as 8 accumulators of 16x16. K staged in double-buffered 32-wide LDS panels
// (stride 36 floats: conflict-free for the 16x4 A / 4x16 B lane patterns) via
// GLOBAL_LOAD_ASYNC_TO_LDS_B128 (ASYNCcnt), overlapping global traffic with WMMA.
__global__ __launch_bounds__(256) void gemm128_wmma(const float* __restrict__ X,
                                                    const float* __restrict__ W,
                                                    const float* __restrict__ bias,
                                                    float* __restrict__ Y, int N) {
    __shared__ float Xs[2][128 * 36];
    __shared__ float Ws[2][128 * 36];
    const int tid  = threadIdx.x;
    const int wv   = tid >> 5;
    const int lane = tid & 31;
    const int l16  = lane & 15;
    const int hi   = lane >> 4;          // 0: K low half, 1: high half
    const int rowBase = blockIdx.x * 128;

    const uint64_t xb = (uint64_t)(uintptr_t)X;
    const uint64_t wb = (uint64_t)(uintptr_t)W;

    // issue one K-panel (128 rows x 32 k) of X and W as direct-to-LDS async loads.
    // rows >= N are clamped to N-1 (their outputs are never stored) so EXEC stays
    // all-ones and no stale LDS padding is needed.
    auto issue_panel = [&](int kp, int buf) {
#pragma unroll
        for (int i = 0; i < 4; ++i) {
            int idx = tid + i * 256;          // 0..1023 float4 slots
            int r = idx >> 3;                 // 0..127
            int c4 = idx & 7;                 // 0..7
            int grow = rowBase + r;
            if (grow >= N) grow = N - 1;
            uint32_t lx = (uint32_t)(uintptr_t)&Xs[buf][r * 36 + c4 * 4];
            uint32_t lw = (uint32_t)(uintptr_t)&Ws[buf][r * 36 + c4 * 4];
            uint32_t ox = (uint32_t)(((uint32_t)grow * HID + kp * 32 + c4 * 4) * 4u);
            uint32_t ow = (uint32_t)(((uint32_t)r    * HID + kp * 32 + c4 * 4) * 4u);
            asm volatile("global_load_async_to_lds_b128 %0, %1, %2"
                         :: "v"(lx), "v"(ox), "s"(xb) : "memory");
            asm volatile("global_load_async_to_lds_b128 %0, %1, %2"
                         :: "v"(lw), "v"(ow), "s"(wb) : "memory");
        }
    };

    v8f acc[8] = {};

    issue_panel(0, 0);
    asm volatile("s_wait_asynccnt 0x0" ::: "memory");
    __syncthreads();

    for (int kp = 0; kp < 4; ++kp) {         // K panels of 32
        const int buf = kp & 1;
        if (kp < 3) issue_panel(kp + 1, buf ^ 1);

        const float* xs = Xs[buf];
        const float* ws = Ws[buf];
        const int arow = (wv * 16 + l16) * 36;
#pragma unroll
        for (int k = 0; k < 32; k += 4) {
            // A (16x4): VGPR0 = {K=0 | K=2}, VGPR1 = {K=1 | K=3} per ISA layout
            v2f a = *(const v2f*)(xs + arow + k + 2 * hi);
#pragma unroll
            for (int ct = 0; ct < 8; ++ct) {
                v2f b = *(const v2f*)(ws + (ct * 16 + l16) * 36 + k + 2 * hi);
                acc[ct] = __builtin_amdgcn_wmma_f32_16x16x4_f32(
                    false, a, false, b, (short)0, acc[ct], false, false);
            }
        }
        if (kp < 3) {
            asm volatile("s_wait_asynccnt 0x0" ::: "memory");
            __syncthreads();
        }
    }

    // C/D layout: VGPR r -> (M=r, lanes 0-15) / (M=r+8, lanes 16-31)
#pragma unroll
    for (int ct = 0; ct < 8; ++ct) {
        int ocol = ct * 16 + l16;
        float bv = bias ? bias[ocol] : 0.0f;
#pragma unroll
        for (int r = 0; r < 8; ++r) {
            int orow = rowBase + wv * 16 + (hi ? r + 8 : r);
            if (orow < N)
                Y[(size_t)orow * HID + ocol] = acc[ct][r] + bv;
        }
    }
}

// ------------------------------------------------------------- edge phase
__global__ __launch_bounds__(256) void edge_logit_max(const float* __restrict__ Q,
                                                      const float* __restrict__ K,
                                                      const float* __restrict__ prio,
                                                      const int* __restrict__ ei, int E,
                                                      float* __restrict__ logits,
                                                      float* __restrict__ mbuf) {
    int i = blockIdx.x * 256 + threadIdx.x;
    if (i >= E * 2) return;
    int e = i >> 1, h = i & 1;
    int si = ei[e], di = ei[E + e];
    const float4* q4 = (const float4*)(Q + (size_t)di * HID + h * 64);
    const float4* k4 = (const float4*)(K + (size_t)si * HID + h * 64);
    float acc = 0.f;
#pragma unroll
    for (int j = 0; j < 16; ++j) {
        float4 a = q4[j], b = k4[j];
        acc += a.x * b.x + a.y * b.y + a.z * b.z + a.w * b.w;
    }
    float lg = acc * prio[h] * 0.125f;   // 1/sqrt(64)
    logits[i] = lg;
    atomicMaxFloat(&mbuf[di * 2 + h], lg);
}

__global__ __launch_bounds__(256) void edge_exp_sum(const int* __restrict__ ei, int E,
                                                    const float* __restrict__ mbuf,
                                                    float* __restrict__ wlog,
                                                    float* __restrict__ zbuf) {
    int i = blockIdx.x * 256 + threadIdx.x;
    if (i >= E * 2) return;
    int e = i >> 1, h = i & 1;
    int di = ei[E + e];
    float w = expf(wlog[i] - mbuf[di * 2 + h]);
    wlog[i] = w;
    atomicAdd(&zbuf[di * 2 + h], w);
}

__global__ __launch_bounds__(256) void edge_aggregate(const int* __restrict__ ei, int E,
                                                      const float* __restrict__ wlog,
                                                      const float* __restrict__ zbuf,
                                                      const float* __restrict__ Vp,
                                                      float* __restrict__ agg) {
    size_t i = (size_t)blockIdx.x * 256 + threadIdx.x;
    if (i >= (size_t)E * HID) return;
    int e = (int)(i >> 7);
    int c = (int)(i & 127);
    int h = c >> 6;
    int si = ei[e], di = ei[E + e];
    float alpha = wlog[e * 2 + h] / (zbuf[di * 2 + h] + 1e-16f);
    atomicAdd(&agg[(size_t)di * HID + c], Vp[(size_t)si * HID + c] * alpha);
}

// ------------------------------------------------------------- epilogues
__global__ __launch_bounds__(256) void gelu_ip(float* __restrict__ x, size_t n) {
    size_t i = (size_t)blockIdx.x * 256 + threadIdx.x;
    if (i >= n) return;
    float v = x[i];
    x[i] = 0.5f * v * (1.0f + erff(v * 0.70710678118654752f));
}

__global__ __launch_bounds__(256) void blend_skip(float* __restrict__ out,
                                                  const float* __restrict__ xbn,
                                                  const float* __restrict__ skip, size_t n) {
    size_t i = (size_t)blockIdx.x * 256 + threadIdx.x;
    if (i >= n) return;
    float a = 1.0f / (1.0f + expf(-skip[0]));
    out[i] = a * out[i] + (1.0f - a) * xbn[i];
}

// ---------------------------------------------------------------- driver
extern "C" void kernel_launch(void* const* d_in, const int* in_sizes, int n_in,
                              void* d_out, int out_size, void* d_ws, size_t ws_size,
                              hipStream_t stream) {
    (void)n_in; (void)out_size; (void)ws_size;
    auto F = [&](int i) { return (const float*)d_in[i]; };
    auto I = [&](int i) { return (const int*)d_in[i]; };

    const int Nu = in_sizes[0] / HID;   // 50000
    const int Ni = in_sizes[1] / HID;   // 100000
    const int Nt = in_sizes[2] / HID;
    const int Nn = in_sizes[3] / HID;
    const int Nm = in_sizes[4] / HID;
    int maxNsrc = Ni;
    if (Nt > maxNsrc) maxNsrc = Nt;
    if (Nn > maxNsrc) maxNsrc = Nn;
    if (Nm > maxNsrc) maxNsrc = Nm;
    if (Nu > maxNsrc) maxNsrc = Nu;

    // params leaves: jax pytree dict-flatten (sorted keys)
    //  bn: item{b,g}, user{b,g}
    const int BN_ITEM_B = 10, BN_ITEM_G = 11, BN_USER_B = 12, BN_USER_G = 13;
    //  node (image,intention,item,taste,user) x {Wa,Wk,Wq,Wv,ba,bk,bq,bv,skip}
    const int ND_IMAGE = 14, ND_INTENT = 23, ND_ITEM = 32, ND_TASTE = 41, ND_USER = 50;
    //  rel (bought_by,buys,im_assoc,n_assoc,t_assoc) x {a,m,p}
    const int RL_BOUGHT = 59, RL_BUYS = 62, RL_IM = 65, RL_N = 68, RL_T = 71;

    char* wp = (char*)d_ws;
    auto alloc = [&](size_t nf) -> float* {
        float* r = (float*)wp;
        wp += ((nf * sizeof(float) + 255) / 256) * 256;
        return r;
    };
    float* bnsum = alloc(512);                       // user[0:256], item[256:512]
    float* coefU = alloc(256);
    float* coefI = alloc(256);
    float* xbnU  = alloc((size_t)Nu * HID);
    float* xbnI  = alloc((size_t)Ni * HID);
    float* Qu    = alloc((size_t)Nu * HID);
    float* Qi    = alloc((size_t)Ni * HID);
    float* WeffK = alloc(HID * HID);
    float* WeffV = alloc(HID * HID);
    float* beffK = alloc(HID);
    float* beffV = alloc(HID);
    float* Kp    = alloc((size_t)maxNsrc * HID);
    float* Vp    = alloc((size_t)maxNsrc * HID);
    float* mbuf  = alloc((size_t)Ni * 2);
    float* zbuf  = alloc((size_t)Ni * 2);
    float* wlog  = alloc((size_t)(in_sizes[8] / 2) * 2 + 1024);  // largest E * H
    float* aggU  = alloc((size_t)Nu * HID);
    float* aggI  = alloc((size_t)Ni * HID);

    // ---- BatchNorm (train stats, biased var)
    hipMemsetAsync(bnsum, 0, 512 * sizeof(float), stream);
    bn_stats<<<(Nu + 63) / 64, 256, 0, stream>>>(F(0), Nu, bnsum);
    bn_stats<<<(Ni + 63) / 64, 256, 0, stream>>>(F(1), Ni, bnsum + 256);
    bn_coef<<<1, 128, 0, stream>>>(bnsum,       Nu, F(BN_USER_G), F(BN_USER_B), coefU);
    bn_coef<<<1, 128, 0, stream>>>(bnsum + 256, Ni, F(BN_ITEM_G), F(BN_ITEM_B), coefI);
    bn_apply<<<(int)(((size_t)Nu * HID + 255) / 256), 256, 0, stream>>>(F(0), coefU, xbnU, Nu);
    bn_apply<<<(int)(((size_t)Ni * HID + 255) / 256), 256, 0, stream>>>(F(1), coefI, xbnI, Ni);

    // ---- Q projections (dst node types only)
    gemm128_wmma<<<(Nu + 127) / 128, 256, 0, stream>>>(xbnU, F(ND_USER + 2), F(ND_USER + 6), Qu, Nu);
    gemm128_wmma<<<(Ni + 127) / 128, 256, 0, stream>>>(xbnI, F(ND_ITEM + 2), F(ND_ITEM + 6), Qi, Ni);

    hipMemsetAsync(aggU, 0, (size_t)Nu * HID * sizeof(float), stream);
    hipMemsetAsync(aggI, 0, (size_t)Ni * HID * sizeof(float), stream);

    // ---- Edge types: fold rel into K/V projections, then 3-pass seg-softmax
    struct ET { const float* Xsrc; int Nsrc; int nd; int rl; const int* ei; int E;
                const float* Qd; float* aggd; int Ndst; };
    ET ets[5] = {
        { F(2), Nt, ND_TASTE,  RL_T,      I(5), in_sizes[5] / 2, Qi, aggI, Ni },
        { F(3), Nn, ND_INTENT, RL_N,      I(6), in_sizes[6] / 2, Qi, aggI, Ni },
        { F(4), Nm, ND_IMAGE,  RL_IM,     I(7), in_sizes[7] / 2, Qi, aggI, Ni },
        { xbnU, Nu, ND_USER,   RL_BUYS,   I(8), in_sizes[8] / 2, Qi, aggI, Ni },
        { xbnI, Ni, ND_ITEM,   RL_BOUGHT, I(9), in_sizes[9] / 2, Qu, aggU, Nu },
    };
    for (int t = 0; t < 5; ++t) {
        const ET& et = ets[t];
        fold_wmat<<<128, 128, 0, stream>>>(F(et.nd + 1), F(et.nd + 5), F(et.rl + 0), WeffK, beffK);
        fold_wmat<<<128, 128, 0, stream>>>(F(et.nd + 3), F(et.nd + 7), F(et.rl + 1), WeffV, beffV);
        gemm128_wmma<<<(et.Nsrc + 127) / 128, 256, 0, stream>>>(et.Xsrc, WeffK, beffK, Kp, et.Nsrc);
        gemm128_wmma<<<(et.Nsrc + 127) / 128, 256, 0, stream>>>(et.Xsrc, WeffV, beffV, Vp, et.Nsrc);

        int nd2 = et.Ndst * 2;
        fill_f32<<<(nd2 + 255) / 256, 256, 0, stream>>>(mbuf, -INFINITY, nd2);
        hipMemsetAsync(zbuf, 0, (size_t)nd2 * sizeof(float), stream);

        int e2 = et.E * 2;
        edge_logit_max<<<(e2 + 255) / 256, 256, 0, stream>>>(et.Qd, Kp, F(et.rl + 2),
                                                             et.ei, et.E, wlog, mbuf);
        edge_exp_sum<<<(e2 + 255) / 256, 256, 0, stream>>>(et.ei, et.E, mbuf, wlog, zbuf);
        size_t ec = (size_t)et.E * HID;
        edge_aggregate<<<(int)((ec + 255) / 256), 256, 0, stream>>>(et.ei, et.E, wlog, zbuf,
                                                                    Vp, et.aggd);
    }

    // ---- Output head: gelu -> Wa -> skip blend
    gelu_ip<<<(int)(((size_t)Nu * HID + 255) / 256), 256, 0, stream>>>(aggU, (size_t)Nu * HID);
    gelu_ip<<<(int)(((size_t)Ni * HID + 255) / 256), 256, 0, stream>>>(aggI, (size_t)Ni * HID);

    float* outU = (float*)d_out;
    float* outI = outU + (size_t)Nu * HID;
    gemm128_wmma<<<(Nu + 127) / 128, 256, 0, stream>>>(aggU, F(ND_USER + 0), F(ND_USER + 4), outU, Nu);
    gemm128_wmma<<<(Ni + 127) / 128, 256, 0, stream>>>(aggI, F(ND_ITEM + 0), F(ND_ITEM + 4), outI, Ni);
    blend_skip<<<(int)(((size_t)Nu * HID + 255) / 256), 256, 0, stream>>>(outU, xbnU, F(ND_USER + 8), (size_t)Nu * HID);
    blend_skip<<<(int)(((size_t)Ni * HID + 255) / 256), 256, 0, stream>>>(outI, xbnI, F(ND_ITEM + 8), (size_t)Ni * HID);
}